// GAT_n_tot_56968446214212
// MI455X (gfx1250) — compile-verified
//
#include <hip/hip_runtime.h>
#include <math.h>

// GAT (4-layer, H=9) + JK-concat + graph max-pool + FC for MI455X (gfx1250).
// Edge phase is L2-atomic bound (all hot arrays << 192MB L2); dense transform
// uses V_WMMA_F32_16X16X4_F32 with LDS-staged operands (no predicated loads).

#define HCH 9
#define LAY 4
#define NCLS 2
#define NGRAPH 2000
#define NEG_SLOPE 0.2f
#define JKW (LAY * HCH)          // 36
#define KPAD 12                  // K padded to 3 WMMA chunks of 4
#define ORDERED_NEG_INF (-2139095041)   // f2oi(-inf) = 0x807FFFFF

typedef float v2f __attribute__((ext_vector_type(2)));
typedef float v8f __attribute__((ext_vector_type(8)));

__device__ __forceinline__ int f2oi(float f) {
    int i = __float_as_int(f);
    return i >= 0 ? i : i ^ 0x7FFFFFFF;
}
__device__ __forceinline__ float oi2f(int i) {
    return __int_as_float(i >= 0 ? i : i ^ 0x7FFFFFFF);
}

// ---------------------------------------------------------------------------
// Kernel 1: per-layer dense transform h = x @ W  (WMMA f32 16x16x4),
// plus alpha_src/alpha_dst dot products, plus per-node init of the
// softmax-max / denom / accumulator buffers for the edge phase.
// Block = 256 thr = 8 waves; each wave owns one 16-node tile (128 rows/block).
// A operand staged zero-padded in LDS by a single cooperative coalesced fill.
// ---------------------------------------------------------------------------
__global__ void __launch_bounds__(256) gat_linear_wmma(
    const float* __restrict__ xin, int xstride,
    const float* __restrict__ Wl,            // [9][9] row-major (layer slice)
    const float* __restrict__ a_src,         // [9]
    const float* __restrict__ a_dst,         // [9]
    float* __restrict__ hout,                // [n][9]
    float* __restrict__ as, float* __restrict__ ad,   // [n]
    int* __restrict__ mbuf, float* __restrict__ denom,
    float* __restrict__ acc,                 // [n][9]
    int n)
{
    __shared__ float Wp[KPAD][16];           // B: W zero-padded, rows K, cols N
    __shared__ float Ap[128][KPAD];          // A: 128 input rows, K zero-padded
    __shared__ float tile[8][16][17];        // staged D tiles (17: bank-conflict pad)

    const int tid  = threadIdx.x;
    const int row0blk = blockIdx.x * 128;

    // --- cooperative fills (single barrier) ---
    if (tid < KPAD * 16) {                   // 192 threads fill B
        int k = tid >> 4, c = tid & 15;
        Wp[k][c] = (k < HCH && c < HCH) ? Wl[k * HCH + c] : 0.0f;
    }
    #pragma unroll
    for (int i = 0; i < (128 * KPAD + 255) / 256; ++i) {   // 6 iters fill A
        int elt = tid + i * 256;             // < 1536
        int r   = elt / KPAD;
        int k   = elt - r * KPAD;
        int gr  = row0blk + r;
        Ap[r][k] = (k < HCH && gr < n) ? xin[(size_t)gr * xstride + k] : 0.0f;
    }
    __syncthreads();

    const int wave = tid >> 5;
    const int lane = tid & 31;
    const int hi   = lane >> 4;              // which half of the wave
    const int lm   = lane & 15;

    v8f dacc = {};
    #pragma unroll
    for (int ck = 0; ck < 3; ++ck) {         // K = 0..11 covers H=9 (rest zero)
        // A layout: lanes 0-15 hold K=4c,4c+1 of row M=lane; 16-31 hold 4c+2,4c+3
        const int k0 = ck * 4 + hi * 2;
        v2f a, b;
        a.x = Ap[wave * 16 + lm][k0];
        a.y = Ap[wave * 16 + lm][k0 + 1];
        b.x = Wp[k0][lm];
        b.y = Wp[k0 + 1][lm];
        dacc = __builtin_amdgcn_wmma_f32_16x16x4_f32(
            /*neg_a=*/false, a, /*neg_b=*/false, b,
            /*c_mod=*/(short)0, dacc, /*reuse_a=*/false, /*reuse_b=*/false);
    }

    // D layout: lane L, vgpr r -> D[M = r + 8*(L/16)][N = L%16]
    #pragma unroll
    for (int r = 0; r < 8; ++r)
        tile[wave][r + 8 * hi][lm] = dacc[r];
    __syncthreads();

    if (lane < 16) {
        const int row = row0blk + wave * 16 + lane;
        if (row < n) {
            float hv[HCH];
            float s = 0.0f, t = 0.0f;
            #pragma unroll
            for (int j = 0; j < HCH; ++j) {
                hv[j] = tile[wave][lane][j];
                s += hv[j] * a_src[j];
                t += hv[j] * a_dst[j];
            }
            #pragma unroll
            for (int j = 0; j < HCH; ++j) {
                hout[(size_t)row * HCH + j] = hv[j];
                acc[(size_t)row * HCH + j]  = 0.0f;
            }
            as[row]    = s;
            ad[row]    = t;
            mbuf[row]  = ORDERED_NEG_INF;
            denom[row] = 0.0f;
        }
    }
}

// ---------------------------------------------------------------------------
// Kernel 2: edge pass 1 — attention logits, segment-max by dst
// (ordered-int encoding so atomicMax(int*) works for floats).
// Edge idx in [E, E+n) are the implicit self-loops.
// ---------------------------------------------------------------------------
__global__ void __launch_bounds__(256) gat_edge_max(
    const int* __restrict__ src, const int* __restrict__ dst,
    const float* __restrict__ as, const float* __restrict__ ad,
    int* __restrict__ mbuf, int nE, int n)
{
    const int idx = blockIdx.x * 256 + threadIdx.x;
    if (idx >= nE + n) return;
    int s, d;
    if (idx < nE) { s = src[idx]; d = dst[idx]; } else { s = d = idx - nE; }
    float e = as[s] + ad[d];
    e = e > 0.0f ? e : NEG_SLOPE * e;
    atomicMax(&mbuf[d], f2oi(e));
}

// ---------------------------------------------------------------------------
// Kernel 3: edge pass 2 — fused exp + denominator + weighted aggregation.
//   acc[d] += w * h[s];  denom[d] += w;   (normalize per-dst in finalize)
// ---------------------------------------------------------------------------
__global__ void __launch_bounds__(256) gat_edge_acc(
    const int* __restrict__ src, const int* __restrict__ dst,
    const float* __restrict__ as, const float* __restrict__ ad,
    const int* __restrict__ mbuf, const float* __restrict__ h,
    float* __restrict__ denom, float* __restrict__ acc, int nE, int n)
{
    const int idx = blockIdx.x * 256 + threadIdx.x;
    if (idx >= nE + n) return;
    int s, d;
    if (idx < nE) { s = src[idx]; d = dst[idx]; } else { s = d = idx - nE; }
    float e = as[s] + ad[d];
    e = e > 0.0f ? e : NEG_SLOPE * e;
    const float w = __expf(e - oi2f(mbuf[d]));
    atomicAdd(&denom[d], w);
    const float* hs = h   + (size_t)s * HCH;
    float*       ac = acc + (size_t)d * HCH;
    #pragma unroll
    for (int j = 0; j < HCH; ++j)
        atomicAdd(&ac[j], w * hs[j]);
}

// ---------------------------------------------------------------------------
// Kernel 4: per-node finalize — normalize, bias, relu, write JK column block,
// and fold the graph max-pool directly in (ordered-int atomicMax into pooled).
// ---------------------------------------------------------------------------
__global__ void __launch_bounds__(256) gat_finalize(
    const float* __restrict__ acc, const float* __restrict__ denom,
    const float* __restrict__ bias, const int* __restrict__ batch,
    float* __restrict__ jk, int* __restrict__ pooled, int layer, int n)
{
    const int nd = blockIdx.x * 256 + threadIdx.x;
    if (nd >= n) return;
    const float inv = 1.0f / denom[nd];       // denom >= exp(0)=1 via self-loop
    const int g = batch[nd];
    #pragma unroll
    for (int j = 0; j < HCH; ++j) {
        float r = fmaf(acc[(size_t)nd * HCH + j], inv, bias[j]);
        r = fmaxf(r, 0.0f);
        jk[(size_t)nd * JKW + layer * HCH + j] = r;
        atomicMax(&pooled[g * JKW + layer * HCH + j], f2oi(r));
    }
}

__global__ void __launch_bounds__(256) gat_init_pooled(int* __restrict__ pooled, int tot)
{
    const int i = blockIdx.x * 256 + threadIdx.x;
    if (i < tot) pooled[i] = ORDERED_NEG_INF;
}

// ---------------------------------------------------------------------------
// Kernel 5: pooled @ fc_w + fc_b  (G*C = 4000 outputs; empty graphs -> 0).
// ---------------------------------------------------------------------------
__global__ void __launch_bounds__(256) gat_fc(
    const int* __restrict__ pooled, const float* __restrict__ fw,
    const float* __restrict__ fb, float* __restrict__ out, int tot)
{
    const int i = blockIdx.x * 256 + threadIdx.x;
    if (i >= tot) return;
    const int g = i / NCLS, c = i % NCLS;
    float sum = fb[c];
    #pragma unroll
    for (int k = 0; k < JKW; ++k) {
        float v = fmaxf(oi2f(pooled[g * JKW + k]), 0.0f);  // -inf (empty graph) -> 0
        sum = fmaf(v, fw[k * NCLS + c], sum);
    }
    out[i] = sum;
}

// ---------------------------------------------------------------------------
extern "C" void kernel_launch(void* const* d_in, const int* in_sizes, int n_in,
                              void* d_out, int out_size, void* d_ws, size_t ws_size,
                              hipStream_t stream) {
    const float* x       = (const float*)d_in[0];
    const int*   ei      = (const int*)  d_in[1];
    const int*   batch   = (const int*)  d_in[2];
    const float* W       = (const float*)d_in[3];
    const float* att_src = (const float*)d_in[4];
    const float* att_dst = (const float*)d_in[5];
    const float* bias    = (const float*)d_in[6];
    const float* fc_w    = (const float*)d_in[7];
    const float* fc_b    = (const float*)d_in[8];
    float*       out     = (float*)d_out;

    const int n  = in_sizes[0] / HCH;        // 200000
    const int E  = in_sizes[1] / 2;          // 6400000
    const int* src = ei;
    const int* dst = ei + E;

    // Workspace carve-up (~47 MB total)
    float* h     = (float*)d_ws;                 // [n][9]
    float* as    = h    + (size_t)n * HCH;       // [n]
    float* ad    = as   + n;                     // [n]
    int*   mbuf  = (int*)(ad + n);               // [n]
    float* denom = (float*)(mbuf + n);           // [n]
    float* acc   = denom + n;                    // [n][9]
    float* jk    = acc  + (size_t)n * HCH;       // [n][36]
    int* pooled  = (int*)(jk + (size_t)n * JKW); // [G][36]

    const dim3 blk(256);
    const int linBlocks  = (n + 127) / 128;      // 128 rows per block
    const int edgeBlocks = (E + n + 255) / 256;
    const int nodeBlocks = (n + 255) / 256;

    gat_init_pooled<<<(NGRAPH * JKW + 255) / 256, blk, 0, stream>>>(pooled, NGRAPH * JKW);

    for (int l = 0; l < LAY; ++l) {
        const float* xin = (l == 0) ? x : (jk + (size_t)(l - 1) * HCH);
        const int xstride = (l == 0) ? HCH : JKW;
        gat_linear_wmma<<<linBlocks, blk, 0, stream>>>(
            xin, xstride, W + l * HCH * HCH, att_src + l * HCH, att_dst + l * HCH,
            h, as, ad, mbuf, denom, acc, n);
        gat_edge_max<<<edgeBlocks, blk, 0, stream>>>(src, dst, as, ad, mbuf, E, n);
        gat_edge_acc<<<edgeBlocks, blk, 0, stream>>>(src, dst, as, ad, mbuf, h, denom, acc, E, n);
        gat_finalize<<<nodeBlocks, blk, 0, stream>>>(acc, denom, bias + l * HCH, batch,
                                                     jk, pooled, l, n);
    }
    gat_fc<<<(NGRAPH * NCLS + 255) / 256, blk, 0, stream>>>(pooled, fc_w, fc_b, out, NGRAPH * NCLS);
}